// SemanticModule_14053132993277
// MI455X (gfx1250) — compile-verified
//
#include <hip/hip_runtime.h>

typedef __attribute__((ext_vector_type(2))) float v2f;
typedef __attribute__((ext_vector_type(8))) float v8f;

#define THREADS 256

// ---------------- elementwise helpers ----------------

__global__ void fill_f32(float* __restrict__ p, float v, int n) {
  int i = blockIdx.x * blockDim.x + threadIdx.x;
  if (i < n) p[i] = v;
}

// pad [n,6] row-major into [n,32] with zero columns 6..31
__global__ void pad_x(const float* __restrict__ x, float* __restrict__ xp, int n32) {
  int i = blockIdx.x * blockDim.x + threadIdx.x;
  if (i >= n32) return;
  int row = i >> 5, col = i & 31;
  xp[i] = (col < 6) ? x[row * 6 + col] : 0.0f;
}

__device__ __forceinline__ void atomAddF(float* p, float v) {
  __hip_atomic_fetch_add(p, v, __ATOMIC_RELAXED, __HIP_MEMORY_SCOPE_AGENT);
}
__device__ __forceinline__ void atomMaxF(float* p, float v) {
  __hip_atomic_fetch_max(p, v, __ATOMIC_RELAXED, __HIP_MEMORY_SCOPE_AGENT);
}

// one atomic per edge into per-destination count (edge lists are constant -> run once)
__global__ void count_edges(const int* __restrict__ dst, float* __restrict__ cnt, int E) {
  int e = blockIdx.x * blockDim.x + threadIdx.x;
  if (e < E) atomAddF(cnt + dst[e], 1.0f);
}

// edge-parallel aggregation: thread = (edge, 4-feature chunk); native f32 atomics
__global__ void agg_edges(const int* __restrict__ ei, int E,
                          const float* __restrict__ xsrc, int K,
                          float* __restrict__ agg, int lc, int isMax) {
  int t = blockIdx.x * blockDim.x + threadIdx.x;
  int e = t >> lc;
  if (e >= E) return;
  int c = (t & ((1 << lc) - 1)) << 2;
  int s = ei[e];
  int d = ei[E + e];
  const float* sp = xsrc + ((long long)s << 5) + c;
  float* dp = agg + ((long long)d << 5) + c;
#pragma unroll
  for (int j = 0; j < 4; ++j) {
    if (c + j < K) {
      float v = sp[j];
      if (isMax) atomMaxF(dp + j, v);
      else       atomAddF(dp + j, v);
    }
  }
}

// mean = sum / max(cnt,1); max: -inf (empty segment) -> 0 like PyG
__global__ void finalize_aggs(float* __restrict__ aggMean, const float* __restrict__ cnt,
                              float* __restrict__ aggMax, int n32) {
  int i = blockIdx.x * blockDim.x + threadIdx.x;
  if (i >= n32) return;
  int row = i >> 5;
  aggMean[i] *= 1.0f / fmaxf(cnt[row], 1.0f);
  float v = aggMax[i];
  aggMax[i] = (v == -__builtin_inff()) ? 0.0f : v;
}

// ---------------- per-layer weight packing ----------------
// Pack each KxN weight (zero-padded to 32x32) into WMMA B-fragment order:
//   packed[((step*2 + tile)*32 + lane)*2 + j] = W[k*32 + n]
//   k = step*4 + 2*(lane>>4) + j ,  n = tile*16 + (lane&15)
// Matrices 0..4 = Wrel[r]; 5 = combined root (rels 0,1,4); 6 = combined root (rels 2,3).
// Also emits combined biases.
__global__ void pack_weights(const float* __restrict__ Wrel, const float* __restrict__ Wroot,
                             const float* __restrict__ brel, int K,
                             float* __restrict__ pk /* 7*1024 */,
                             float* __restrict__ bcS, float* __restrict__ bcB) {
  int i = blockIdx.x * blockDim.x + threadIdx.x;
  int KN = K * 32;
  if (i < 7 * 1024) {
    int m   = i >> 10;
    int r   = i & 1023;
    int j    = r & 1;
    int lane = (r >> 1) & 31;
    int tile = (r >> 6) & 1;
    int step = r >> 7;
    int k = step * 4 + 2 * (lane >> 4) + j;
    int n = tile * 16 + (lane & 15);
    float v = 0.0f;
    if (k < K) {
      int o = k * 32 + n;
      if (m < 5)       v = Wrel[m * KN + o];
      else if (m == 5) v = Wroot[o] + Wroot[KN + o] + Wroot[4 * KN + o];
      else             v = Wroot[2 * KN + o] + Wroot[3 * KN + o];
    }
    pk[i] = v;
  } else if (i < 7 * 1024 + 32) {
    int n = i - 7 * 1024;
    bcS[n] = brel[n] + brel[32 + n] + brel[4 * 32 + n];
  } else if (i < 7 * 1024 + 64) {
    int n = i - (7 * 1024 + 32);
    bcB[n] = brel[2 * 32 + n] + brel[3 * 32 + n];
  }
}

// ---------------- fused linear via V_WMMA_F32_16X16X4_F32 ----------------
// One wave -> 16 output rows x 32 cols (two 16x16 f32 accumulators), K fixed = 32.
// All A matrices are stride-32 zero-padded; B matrices pre-packed in fragment order.
// A frag: lane L, vgpr j holds (M = L%16, K = kb + 2*(L/16)+j)  -> one v2f load
// C/D:    lane L, vgpr v holds (M = v + 8*(L/16), N = L%16)
__global__ void __launch_bounds__(256)
wmma_linear(const float* __restrict__ A0, const float* __restrict__ pB0,
            const float* __restrict__ A1, const float* __restrict__ pB1,
            const float* __restrict__ A2, const float* __restrict__ pB2,
            const float* __restrict__ A3, const float* __restrict__ pB3,
            const float* __restrict__ bias, const float* __restrict__ resid,
            float* __restrict__ out, int nrows, int relu) {
  int wave = blockIdx.x * (blockDim.x >> 5) + (threadIdx.x >> 5);
  int row0 = wave * 16;
  if (row0 >= nrows) return;

  int lane = threadIdx.x & 31;
  int half = lane >> 4;
  int lm   = lane & 15;

  v8f c0 = {};
  v8f c1 = {};

  const float* As[4] = {A0, A1, A2, A3};
  const float* Bs[4] = {pB0, pB1, pB2, pB3};

  for (int m = 0; m < 4; ++m) {
    const float* A = As[m];
    if (!A) continue;
    const float* Ap = A + (((long long)row0 + lm) << 5) + 2 * half;
    const float* Bp = Bs[m] + 2 * lane;
#pragma unroll
    for (int step = 0; step < 8; ++step) {
      v2f a  = *(const v2f*)(Ap + step * 4);
      v2f b0 = *(const v2f*)(Bp + (step * 2 + 0) * 64);
      v2f b1 = *(const v2f*)(Bp + (step * 2 + 1) * 64);
      c0 = __builtin_amdgcn_wmma_f32_16x16x4_f32(false, a, false, b0, (short)0, c0, false, false);
      c1 = __builtin_amdgcn_wmma_f32_16x16x4_f32(false, a, false, b1, (short)0, c1, false, false);
    }
  }

  float bs0 = bias[lm];
  float bs1 = bias[16 + lm];
#pragma unroll
  for (int v = 0; v < 8; ++v) {
    int row = row0 + v + 8 * half;
    long long base = (long long)row << 5;
    float o0 = c0[v] + bs0;
    float o1 = c1[v] + bs1;
    if (resid) { o0 += resid[base + lm]; o1 += resid[base + 16 + lm]; }
    if (relu)  { o0 = fmaxf(o0, 0.0f);   o1 = fmaxf(o1, 0.0f); }
    out[base + lm]      = o0;
    out[base + 16 + lm] = o1;
  }
}

// ---------------- host orchestration ----------------

static inline int nblocks(long long n) { return (int)((n + THREADS - 1) / THREADS); }

extern "C" void kernel_launch(void* const* d_in, const int* in_sizes, int n_in,
                              void* d_out, int out_size, void* d_ws, size_t ws_size,
                              hipStream_t stream) {
  (void)n_in; (void)out_size; (void)ws_size;

  const float* x_stroke = (const float*)d_in[0];
  const float* x_brep   = (const float*)d_in[1];
  const int* ei[5] = {(const int*)d_in[2], (const int*)d_in[3], (const int*)d_in[4],
                      (const int*)d_in[5], (const int*)d_in[6]};
  const float* W0_rel  = (const float*)d_in[7];
  const float* b0_rel  = (const float*)d_in[8];
  const float* W0_root = (const float*)d_in[9];
  const float* Wb_rel  = (const float*)d_in[10];
  const float* bb_rel  = (const float*)d_in[11];
  const float* Wb_root = (const float*)d_in[12];

  const int NS = in_sizes[0] / 6;   // 100000 stroke nodes (multiple of 16)
  const int NB = in_sizes[1] / 6;   // 50000 brep nodes (multiple of 16)
  const int E  = in_sizes[2] / 2;   // 1,600,000 edges per relation

  // workspace layout (floats)
  float* ws = (float*)d_ws;
  size_t off = 0;
  float* xsP  = ws + off; off += (size_t)NS * 32;   // padded stroke input
  float* xbP  = ws + off; off += (size_t)NB * 32;   // padded brep input
  float* xsA  = ws + off; off += (size_t)NS * 32;
  float* xbA  = ws + off; off += (size_t)NB * 32;
  float* xsB  = ws + off; off += (size_t)NS * 32;
  float* xbB  = ws + off; off += (size_t)NB * 32;
  float* aggS0 = ws + off; off += (size_t)NS * 32;  // rel0 mean-sum  (stroke)
  float* aggS1 = ws + off; off += (size_t)NS * 32;  // rel1 add       (stroke)
  float* aggS4 = ws + off; off += (size_t)NS * 32;  // rel4 max       (stroke)
  float* aggB2 = ws + off; off += (size_t)NB * 32;  // rel2 mean-sum  (brep)
  float* aggB3 = ws + off; off += (size_t)NB * 32;  // rel3 max       (brep)
  float* cnt0  = ws + off; off += (size_t)NS;
  float* cnt2  = ws + off; off += (size_t)NB;
  float* pk    = ws + off; off += 7 * 1024;         // packed weights (frag order)
  float* bcS   = ws + off; off += 32;
  float* bcB   = ws + off; off += 32;

  // one-time per launch: pad inputs, destination counts
  pad_x<<<nblocks((long long)NS * 32), THREADS, 0, stream>>>(x_stroke, xsP, NS * 32);
  pad_x<<<nblocks((long long)NB * 32), THREADS, 0, stream>>>(x_brep, xbP, NB * 32);
  hipMemsetAsync(cnt0, 0, (size_t)NS * sizeof(float), stream);
  hipMemsetAsync(cnt2, 0, (size_t)NB * sizeof(float), stream);
  count_edges<<<nblocks(E), THREADS, 0, stream>>>(ei[0] + E, cnt0, E);
  count_edges<<<nblocks(E), THREADS, 0, stream>>>(ei[2] + E, cnt2, E);

  const float* xs = xsP;
  const float* xb = xbP;

  for (int layer = 0; layer < 5; ++layer) {
    const int K  = (layer == 0) ? 6 : 32;
    const int lc = (layer == 0) ? 1 : 3;   // log2(feature chunks per edge)
    const float* Wrel  = (layer == 0) ? W0_rel  : Wb_rel  + (size_t)(layer - 1) * 5 * 32 * 32;
    const float* brel  = (layer == 0) ? b0_rel  : bb_rel  + (size_t)(layer - 1) * 5 * 32;
    const float* Wroot = (layer == 0) ? W0_root : Wb_root + (size_t)(layer - 1) * 5 * 32 * 32;

    pack_weights<<<nblocks(7 * 1024 + 64), THREADS, 0, stream>>>(Wrel, Wroot, brel, K,
                                                                 pk, bcS, bcB);

    hipMemsetAsync(aggS0, 0, (size_t)NS * 32 * sizeof(float), stream);
    hipMemsetAsync(aggS1, 0, (size_t)NS * 32 * sizeof(float), stream);
    hipMemsetAsync(aggB2, 0, (size_t)NB * 32 * sizeof(float), stream);
    fill_f32<<<nblocks((long long)NS * 32), THREADS, 0, stream>>>(aggS4, -__builtin_inff(), NS * 32);
    fill_f32<<<nblocks((long long)NB * 32), THREADS, 0, stream>>>(aggB3, -__builtin_inff(), NB * 32);

    const long long tE = (long long)E << lc;
    agg_edges<<<nblocks(tE), THREADS, 0, stream>>>(ei[0], E, xs, K, aggS0, lc, 0); // mean-sum
    agg_edges<<<nblocks(tE), THREADS, 0, stream>>>(ei[1], E, xs, K, aggS1, lc, 0); // add
    agg_edges<<<nblocks(tE), THREADS, 0, stream>>>(ei[2], E, xs, K, aggB2, lc, 0); // mean-sum
    agg_edges<<<nblocks(tE), THREADS, 0, stream>>>(ei[3], E, xb, K, aggB3, lc, 1); // max
    agg_edges<<<nblocks(tE), THREADS, 0, stream>>>(ei[4], E, xs, K, aggS4, lc, 1); // max

    finalize_aggs<<<nblocks((long long)NS * 32), THREADS, 0, stream>>>(aggS0, cnt0, aggS4, NS * 32);
    finalize_aggs<<<nblocks((long long)NB * 32), THREADS, 0, stream>>>(aggB2, cnt2, aggB3, NB * 32);

    float* outS; float* outB;
    if (layer == 4)            { outS = (float*)d_out; outB = (float*)d_out + (size_t)NS * 32; }
    else if ((layer & 1) == 0) { outS = xsA; outB = xbA; }
    else                       { outS = xsB; outB = xbB; }
    const float* residS = (layer == 0) ? nullptr : xs;
    const float* residB = (layer == 0) ? nullptr : xb;
    const int relu = (layer == 4) ? 1 : 0;

    const int wavesS = NS / 16;
    wmma_linear<<<(wavesS + 7) / 8, 256, 0, stream>>>(
        aggS0, pk + 0 * 1024,
        aggS1, pk + 1 * 1024,
        aggS4, pk + 4 * 1024,
        xs,    pk + 5 * 1024,
        bcS, residS, outS, NS, relu);

    const int wavesB = NB / 16;
    wmma_linear<<<(wavesB + 7) / 8, 256, 0, stream>>>(
        aggB2, pk + 2 * 1024,
        aggB3, pk + 3 * 1024,
        nullptr, nullptr,
        xb,    pk + 6 * 1024,
        bcB, residB, outB, NB, relu);

    xs = outS;
    xb = outB;
  }
}